// TimeMixingModule_4234837754289
// MI455X (gfx1250) — compile-verified
//
#include <hip/hip_runtime.h>
#include <math.h>

// MI455X / gfx1250: wave32, WMMA-based fp32 GEMM (V_WMMA_F32_16X16X4_F32).
// Fused RWKV graph time-mixing:
//   Stage 1 (x3): node projections K/V/R with time-mix + activation fused,
//                 node-level (activations commute with the edge gather).
//   Stage 2:      per-edge wkv computation fused into the output GEMM vs Wo.

typedef float v2f __attribute__((ext_vector_type(2)));
typedef float v8f __attribute__((ext_vector_type(8)));

#define C 128
#define WAVES 8
#define ROWS_PER_BLOCK (WAVES * 16)

__device__ __forceinline__ v8f wmma_f32(v2f a, v2f b, v8f c) {
  // D = A(16x4 f32) * B(4x16 f32) + C(16x16 f32)
  return __builtin_amdgcn_wmma_f32_16x16x4_f32(
      /*neg_a=*/false, a, /*neg_b=*/false, b,
      /*c_mod=*/(short)0, c, /*reuse_a=*/false, /*reuse_b=*/false);
}

// Stage a 128x128 row-major weight matrix into LDS with an XOR swizzle:
// physical col = logical col ^ ((row & 15) << 2). This makes the B-fragment
// read pattern (16 consecutive rows, float2 at col k0 + 2h) span all 64 LDS
// banks conflict-free while keeping LDS at exactly 64 KB.
__device__ __forceinline__ void load_w_lds(float* Wlds, const float* __restrict__ W, int tid) {
#pragma unroll
  for (int g = 0; g < 16; ++g) {
    int f4  = g * 256 + tid;          // float4 index, fully coalesced global read
    int row = f4 >> 5;                // 32 float4s per row
    int c4  = (f4 & 31) << 2;
    int scol = c4 ^ ((row & 15) << 2);
    float4 v = *reinterpret_cast<const float4*>(W + row * C + c4);
    *reinterpret_cast<float4*>(Wlds + row * C + scol) = v;
  }
}

// One 16x16 output tile: out[m][n0+n] = sum_c A[m][c] * W[n0+n][c]
// A-fragments for all 32 K-steps live in registers (afr). B-fragments come
// from swizzled LDS as ds_load_b64. Two accumulators break the WMMA RAW chain.
__device__ __forceinline__ v8f gemm_tile(const v2f* afr, const float* Wlds,
                                         int n0, int r, int h) {
  v8f acc0 = {0.f, 0.f, 0.f, 0.f, 0.f, 0.f, 0.f, 0.f};
  v8f acc1 = {0.f, 0.f, 0.f, 0.f, 0.f, 0.f, 0.f, 0.f};
  int wrow = n0 + r;                      // B lane row = W row n0 + (lane&15)
  int swz  = (wrow & 15) << 2;
  const float* wr = Wlds + wrow * C + 2 * h;
#pragma unroll
  for (int t = 0; t < 32; t += 2) {
    v2f b0 = *reinterpret_cast<const v2f*>(wr + ((4 * t) ^ swz));
    acc0 = wmma_f32(afr[t], b0, acc0);
    v2f b1 = *reinterpret_cast<const v2f*>(wr + ((4 * (t + 1)) ^ swz));
    acc1 = wmma_f32(afr[t + 1], b1, acc1);
  }
  return acc0 + acc1;
}

// MODE: 0 = K (exp(min(.,60))), 1 = V (identity), 2 = R (sigmoid)
template <int MODE>
__global__ void __launch_bounds__(256) node_proj(const float* __restrict__ x,
                                                 const float* __restrict__ px,
                                                 const float* __restrict__ W,
                                                 const float* __restrict__ tmix,
                                                 float* __restrict__ out, int N) {
  __shared__ float Wlds[C * C];
  int tid = threadIdx.x;
  load_w_lds(Wlds, W, tid);
  __syncthreads();

  int wave = tid >> 5, lane = tid & 31;
  int r = lane & 15, h = lane >> 4;
  int m0  = (blockIdx.x * WAVES + wave) * 16;
  int row = m0 + r;

  // A-fragments in native 16x4 f32 WMMA layout: lane holds row m0+(lane&15),
  // float2 at column 4t + 2*(lane>>4). Time-mix computed inline.
  v2f afr[32];
  if (row < N) {
    const float* xp = x  + (size_t)row * C + 2 * h;
    const float* pp = px + (size_t)row * C + 2 * h;
    const float* tp = tmix + 2 * h;
#pragma unroll
    for (int t = 0; t < 32; ++t) {
      int c = 4 * t;
      v2f xv = *reinterpret_cast<const v2f*>(xp + c);
      v2f pv = *reinterpret_cast<const v2f*>(pp + c);
      v2f tm = *reinterpret_cast<const v2f*>(tp + c);
      v2f one = {1.0f, 1.0f};
      afr[t] = xv * tm + pv * (one - tm);
    }
  } else {
#pragma unroll
    for (int t = 0; t < 32; ++t) afr[t] = (v2f){0.0f, 0.0f};
  }

#pragma unroll
  for (int j = 0; j < 8; ++j) {
    int n0 = j * 16;
    v8f acc = gemm_tile(afr, Wlds, n0, r, h);
#pragma unroll
    for (int rr = 0; rr < 8; ++rr) {
      int orow = m0 + rr + 8 * h;   // D layout: VGPR rr, lane half h
      if (orow < N) {
        float v = acc[rr];
        if (MODE == 0) v = expf(fminf(v, 60.0f));
        else if (MODE == 2) v = 1.0f / (1.0f + expf(-v));
        out[(size_t)orow * C + n0 + r] = v;
      }
    }
  }
}

__global__ void __launch_bounds__(256) edge_out(const float* __restrict__ Kexp,
                                                const float* __restrict__ Vn,
                                                const float* __restrict__ Rs,
                                                const float* __restrict__ ah,
                                                const float* __restrict__ bh,
                                                const float* __restrict__ Wo,
                                                const float* __restrict__ time_w,
                                                const int* __restrict__ sIdx,
                                                const int* __restrict__ tIdx,
                                                const int* __restrict__ ip,
                                                const int* __restrict__ Tp,
                                                float* __restrict__ out, int E) {
  __shared__ float Wlds[C * C];
  int tid = threadIdx.x;
  load_w_lds(Wlds, Wo, tid);
  __syncthreads();

  int wave = tid >> 5, lane = tid & 31;
  int r = lane & 15, h = lane >> 4;
  int m0   = (blockIdx.x * WAVES + wave) * 16;
  int erow = m0 + r;

  // Build rw = r * (ah + w*k*v) / (bh + k + eps) straight into A-fragments.
  v2f afr[32];
  if (erow < E) {
    int si = sIdx[erow];
    int ti = tIdx[erow];
    float dec = (float)(ip[0] + 1 - Tp[0]);   // -(T - i - 1)
    const float* kp = Kexp + (size_t)si * C + 2 * h;
    const float* vp = Vn   + (size_t)ti * C + 2 * h;
    const float* rp = Rs   + (size_t)si * C + 2 * h;
    const float* ap = ah + (size_t)erow * C + 2 * h;
    const float* bp = bh + (size_t)erow * C + 2 * h;
    const float* wp = time_w + 2 * h;
#pragma unroll
    for (int t = 0; t < 32; ++t) {
      int c = 4 * t;
      v2f ke = *reinterpret_cast<const v2f*>(kp + c);
      v2f vv = *reinterpret_cast<const v2f*>(vp + c);
      v2f rg = *reinterpret_cast<const v2f*>(rp + c);
      v2f aa = *reinterpret_cast<const v2f*>(ap + c);
      v2f bb = *reinterpret_cast<const v2f*>(bp + c);
      v2f tw = *reinterpret_cast<const v2f*>(wp + c);
      v2f w;
      w.x = expf(dec * tw.x);
      w.y = expf(dec * tw.y);
      v2f eps = {1e-20f, 1e-20f};
      v2f num = aa + w * ke * vv;
      v2f den = bb + ke + eps;
      afr[t] = rg * num / den;
    }
  } else {
#pragma unroll
    for (int t = 0; t < 32; ++t) afr[t] = (v2f){0.0f, 0.0f};
  }

#pragma unroll
  for (int j = 0; j < 8; ++j) {
    int n0 = j * 16;
    v8f acc = gemm_tile(afr, Wlds, n0, r, h);
#pragma unroll
    for (int rr = 0; rr < 8; ++rr) {
      int orow = m0 + rr + 8 * h;
      if (orow < E) out[(size_t)orow * C + n0 + r] = acc[rr];
    }
  }
}

extern "C" void kernel_launch(void* const* d_in, const int* in_sizes, int n_in,
                              void* d_out, int out_size, void* d_ws, size_t ws_size,
                              hipStream_t stream) {
  const float* x   = (const float*)d_in[0];
  const float* px  = (const float*)d_in[1];
  const float* ah  = (const float*)d_in[2];
  const float* bh  = (const float*)d_in[3];
  const float* Wk  = (const float*)d_in[4];
  const float* Wv  = (const float*)d_in[5];
  const float* Wr  = (const float*)d_in[6];
  const float* Wo  = (const float*)d_in[7];
  const float* tmk = (const float*)d_in[8];
  const float* tmv = (const float*)d_in[9];
  const float* tmr = (const float*)d_in[10];
  const float* tw  = (const float*)d_in[11];
  const int*   s   = (const int*)d_in[12];
  const int*   t   = (const int*)d_in[13];
  const int*   ip  = (const int*)d_in[14];
  const int*   Tp  = (const int*)d_in[15];

  int N = in_sizes[0] / C;     // 25000
  int E = in_sizes[12];        // 400000

  // Workspace: node-level Kexp / V / Rsig buffers (3 * N * C floats).
  float* Kexp = (float*)d_ws;
  float* Vn   = Kexp + (size_t)N * C;
  float* Rs   = Vn   + (size_t)N * C;

  int nblocksA = (N + ROWS_PER_BLOCK - 1) / ROWS_PER_BLOCK;
  node_proj<0><<<nblocksA, 256, 0, stream>>>(x, px, Wk, tmk, Kexp, N);
  node_proj<1><<<nblocksA, 256, 0, stream>>>(x, px, Wv, tmv, Vn, N);
  node_proj<2><<<nblocksA, 256, 0, stream>>>(x, px, Wr, tmr, Rs, N);

  int nblocksB = (E + ROWS_PER_BLOCK - 1) / ROWS_PER_BLOCK;
  edge_out<<<nblocksB, 256, 0, stream>>>(Kexp, Vn, Rs, ah, bh, Wo, tw,
                                         s, t, ip, Tp, (float*)d_out, E);
}